// SOM_29429115912593
// MI455X (gfx1250) — compile-verified
//
#include <hip/hip_runtime.h>

typedef __attribute__((ext_vector_type(2))) float v2f;
typedef __attribute__((ext_vector_type(4))) float v4f;
typedef __attribute__((ext_vector_type(8))) float v8f;

#define D_K  128     // depth
#define MN_N 4096    // number of SOM units (output cols)
#define B_M  8192    // batch (output rows)
#define BM   128     // block tile M
#define BN   128     // block tile N
#define LDW  132     // padded LDS row stride in floats (128 + 4 -> conflict-free)

__global__ __launch_bounds__(256)
void som_dist_kernel(const float* __restrict__ batch,
                     const float* __restrict__ weights,
                     float* __restrict__ out) {
  extern __shared__ float smem[];
  float* As  = smem;                   // BM x LDW   (batch tile, row-major M x K)
  float* Ws  = smem + BM * LDW;        // BN x LDW   (weights tile, row-major N x K)
  float* a2s = Ws + BN * LDW;          // BM row norms of batch tile
  float* b2s = a2s + BM;               // BN row norms of weights tile

  const int tid   = threadIdx.x;
  const int mBase = blockIdx.y * BM;
  const int nBase = blockIdx.x * BN;

  // ---- Stage tiles into LDS: one full 512B row per wave per iteration ----
  {
    const int rowInit = tid >> 5;        // 0..7
    const int col     = (tid & 31) * 4;  // 0..124
#pragma unroll
    for (int it = 0; it < 16; ++it) {
      const int r = rowInit + it * 8;
      v4f av = *(const v4f*)(batch   + (size_t)(mBase + r) * D_K + col);
      v4f wv = *(const v4f*)(weights + (size_t)(nBase + r) * D_K + col);
      *(v4f*)(As + r * LDW + col) = av;
      *(v4f*)(Ws + r * LDW + col) = wv;
    }
  }
  __syncthreads();

  // ---- Per-row squared norms from LDS (threads 0-127: A rows, 128-255: W rows) ----
  {
    const int rr = tid & 127;
    const float* rowp = (tid < 128) ? (As + rr * LDW) : (Ws + rr * LDW);
    float s = 0.0f;
#pragma unroll
    for (int k = 0; k < D_K; k += 4) {
      v4f v = *(const v4f*)(rowp + k);
      s = fmaf(v.x, v.x, s);
      s = fmaf(v.y, v.y, s);
      s = fmaf(v.z, v.z, s);
      s = fmaf(v.w, v.w, s);
    }
    if (tid < 128) a2s[rr] = s;
    else           b2s[rr] = s;
  }
  __syncthreads();

  // ---- WMMA main loop: 8 waves, each owns a 32(M) x 64(N) sub-tile ----
  const int w    = tid >> 5;        // wave 0..7
  const int lane = tid & 31;
  const int wm   = (w & 3) * 32;    // 4 waves along M
  const int wn   = (w >> 2) * 64;   // 2 waves along N
  const int lrow = lane & 15;       // M (for A frag) / N (for B frag) within 16-tile
  const int lhi  = lane >> 4;       // selects K pair {0,1} vs {2,3}

  // f32 WMMA 16x16x4 fragment: lane L   -> (row L,  K0..K1)  = ds_load_b64
  //                            lane L+16-> (row L,  K2..K3)  = ds_load_b64
  // Identical addressing for A (batch rows, M-major) and B (= weights^T, so
  // weights rows kept row-major [n][k] already give the needed N-major layout).
  const float* aBase = As + (wm + lrow) * LDW + lhi * 2;
  const float* bBase = Ws + (wn + lrow) * LDW + lhi * 2;

  v8f acc[2][4] = {};  // 2 M-tiles x 4 N-tiles of 16x16 f32 accumulators

#pragma unroll
  for (int k = 0; k < D_K; k += 4) {
    v2f a0 = *(const v2f*)(aBase + k);
    v2f a1 = *(const v2f*)(aBase + 16 * LDW + k);
    v2f b0 = *(const v2f*)(bBase + k);
    v2f b1 = *(const v2f*)(bBase + 16 * LDW + k);
    v2f b2 = *(const v2f*)(bBase + 32 * LDW + k);
    v2f b3 = *(const v2f*)(bBase + 48 * LDW + k);
    // 8 args: (neg_a, A, neg_b, B, c_mod, C, reuse_a, reuse_b)
    acc[0][0] = __builtin_amdgcn_wmma_f32_16x16x4_f32(false, a0, false, b0, (short)0, acc[0][0], false, false);
    acc[0][1] = __builtin_amdgcn_wmma_f32_16x16x4_f32(false, a0, false, b1, (short)0, acc[0][1], false, false);
    acc[0][2] = __builtin_amdgcn_wmma_f32_16x16x4_f32(false, a0, false, b2, (short)0, acc[0][2], false, false);
    acc[0][3] = __builtin_amdgcn_wmma_f32_16x16x4_f32(false, a0, false, b3, (short)0, acc[0][3], false, false);
    acc[1][0] = __builtin_amdgcn_wmma_f32_16x16x4_f32(false, a1, false, b0, (short)0, acc[1][0], false, false);
    acc[1][1] = __builtin_amdgcn_wmma_f32_16x16x4_f32(false, a1, false, b1, (short)0, acc[1][1], false, false);
    acc[1][2] = __builtin_amdgcn_wmma_f32_16x16x4_f32(false, a1, false, b2, (short)0, acc[1][2], false, false);
    acc[1][3] = __builtin_amdgcn_wmma_f32_16x16x4_f32(false, a1, false, b3, (short)0, acc[1][3], false, false);
  }

  // ---- Epilogue: out = a2[m] + b2[n] - 2*dot  (C/D layout: VGPR r -> M row,
  //      lanes 0-15 -> N, lanes 16-31 -> M+8) ----
#pragma unroll
  for (int mi = 0; mi < 2; ++mi) {
#pragma unroll
    for (int ni = 0; ni < 4; ++ni) {
      v8f c = acc[mi][ni];
      const int n  = wn + ni * 16 + lrow;
      const int m0 = wm + mi * 16 + lhi * 8;
      const float bn = b2s[n];
      float* op = out + (size_t)(mBase + m0) * MN_N + (nBase + n);
#pragma unroll
      for (int r = 0; r < 8; ++r) {
        const float am = a2s[m0 + r];
        op[(size_t)r * MN_N] = fmaf(c[r], -2.0f, am + bn);
      }
    }
  }
}

extern "C" void kernel_launch(void* const* d_in, const int* in_sizes, int n_in,
                              void* d_out, int out_size, void* d_ws, size_t ws_size,
                              hipStream_t stream) {
  const float* batch   = (const float*)d_in[0];  // (8192, 128) f32
  const float* weights = (const float*)d_in[1];  // (4096, 128) f32
  float* out = (float*)d_out;                    // (8192, 4096) f32

  dim3 grid(MN_N / BN, B_M / BM);  // (32, 64)
  const size_t smem = (size_t)(2 * BM * LDW + BM + BN) * sizeof(float);  // 136192 B
  som_dist_kernel<<<grid, dim3(256), smem, stream>>>(batch, weights, out);
}